// GMaxPool_13640816132108
// MI455X (gfx1250) — compile-verified
//
#include <hip/hip_runtime.h>

// C4 group max-pool with inverse rotations.
// out[b,c,h,w] = max( s0[h,w], s1[63-w,h], s2[63-h,63-w], s3[w,63-h] )
// One block per (b,c) image (64x64 f32). Slices 1/3 are DMA'd into LDS
// transposed via async global->LDS (CDNA5 ASYNCcnt path); slices 0/2 are
// read straight from global as float4 (row / reversed-row = coalesced).

#define TILE_STRIDE 65  // 64 + 1 pad: bank = (row + col) % 64 -> conflict-free

__global__ __launch_bounds__(256) void gmaxpool_c4_kernel(
    const float* __restrict__ x, float* __restrict__ out) {
  const int bc  = blockIdx.x;     // 0 .. 32*64-1
  const int tid = threadIdx.x;    // 0 .. 255

  const float* img = x + (size_t)bc * 4 * 4096;
  const float* s0  = img;              // identity
  const float* s1  = img + 1 * 4096;   // need s1[63-w, h]
  const float* s2  = img + 2 * 4096;   // need s2[63-h, 63-w]
  const float* s3  = img + 3 * 4096;   // need s3[w, 63-h]

  // Transposed tiles: tN[c * TILE_STRIDE + r] = sN[r][c]
  __shared__ float t1[64 * TILE_STRIDE];
  __shared__ float t3[64 * TILE_STRIDE];

  // ---- Async DMA: global -> LDS, transposed scatter (b32 per lane). ----
  // 4096 elements per tile / 256 threads = 16 ops per thread per tile.
  // Global side: each wave op covers 128 contiguous bytes (one line).
  // LDS side: stride-65 layout -> 32 distinct banks per wave op.
  {
    const unsigned lds1 = (unsigned)(size_t)(&t1[0]);
    const unsigned lds3 = (unsigned)(size_t)(&t3[0]);
#pragma unroll
    for (int it = 0; it < 16; ++it) {
      const int idx = tid + it * 256;   // flat element = r*64 + c
      const int r = idx >> 6;
      const int c = idx & 63;
      const unsigned goff = (unsigned)(idx * 4);                    // bytes
      const unsigned loff = (unsigned)((c * TILE_STRIDE + r) * 4);  // bytes
      asm volatile("global_load_async_to_lds_b32 %0, %1, %2"
                   :
                   : "v"(lds1 + loff), "v"(goff), "s"(s1)
                   : "memory");
      asm volatile("global_load_async_to_lds_b32 %0, %1, %2"
                   :
                   : "v"(lds3 + loff), "v"(goff), "s"(s3)
                   : "memory");
    }
  }

  // ---- Overlap: pull slice0 / slice2 data while the DMA runs. ----
  float4 v0[4], v2[4];
#pragma unroll
  for (int k = 0; k < 4; ++k) {
    const int f  = tid + k * 256;   // float4 index: h = f>>4, w4 = f&15
    const int h  = f >> 4;
    const int w4 = f & 15;
    v0[k] = ((const float4*)s0)[f];
    // s2[63-h][60-w .. 63-w] (components reversed at use site)
    v2[k] = ((const float4*)s2)[(63 - h) * 16 + (15 - w4)];
  }

  // ---- Wait for async engine, then make LDS visible to all waves. ----
#if __has_builtin(__builtin_amdgcn_s_wait_asynccnt)
  __builtin_amdgcn_s_wait_asynccnt(0);
#else
  asm volatile("s_wait_asynccnt 0" ::: "memory");
#endif
  __syncthreads();

  // ---- Compute + coalesced float4 stores. ----
  float* outimg = out + (size_t)bc * 4096;
#pragma unroll
  for (int k = 0; k < 4; ++k) {
    const int f = tid + k * 256;
    const int h = f >> 4;
    const int w = (f & 15) << 2;

    // Row-contiguous reads in the transposed tiles (conflict-free).
    const float* t1row = &t1[h * TILE_STRIDE];         // t1row[63-w-j] = s1[63-w-j][h]
    const float* t3row = &t3[(63 - h) * TILE_STRIDE];  // t3row[w+j]    = s3[w+j][63-h]

    float4 r;
    r.x = fmaxf(fmaxf(v0[k].x, v2[k].w), fmaxf(t1row[63 - w], t3row[w + 0]));
    r.y = fmaxf(fmaxf(v0[k].y, v2[k].z), fmaxf(t1row[62 - w], t3row[w + 1]));
    r.z = fmaxf(fmaxf(v0[k].z, v2[k].y), fmaxf(t1row[61 - w], t3row[w + 2]));
    r.w = fmaxf(fmaxf(v0[k].w, v2[k].x), fmaxf(t1row[60 - w], t3row[w + 3]));

    ((float4*)outimg)[f] = r;
  }
}

extern "C" void kernel_launch(void* const* d_in, const int* in_sizes, int n_in,
                              void* d_out, int out_size, void* d_ws, size_t ws_size,
                              hipStream_t stream) {
  const float* x = (const float*)d_in[0];
  float* out = (float*)d_out;
  (void)in_sizes; (void)n_in; (void)out_size; (void)d_ws; (void)ws_size;
  dim3 grid(32 * 64);   // one block per (b, c)
  dim3 block(256);      // 8 wave32
  gmaxpool_c4_kernel<<<grid, block, 0, stream>>>(x, out);
}